// VarianceInputLayer_46875273068798
// MI455X (gfx1250) — compile-verified
//
#include <hip/hip_runtime.h>
#include <stdint.h>

typedef __attribute__((ext_vector_type(2))) float v2f;
typedef __attribute__((ext_vector_type(8))) float v8f;

#define H 512
#define W 512
#define COUT 12

// 512 threads = 16 waves; workgroup stages a 72x72 halo for a 64x64 output
// region; each wave owns one 16x16 sub-tile and a private row-sum buffer.
__global__ __launch_bounds__(512)
void varpool_wmma_kernel(const float* __restrict__ x, float* __restrict__ out) {
    // Halo: rows 0..71 = input region rows -4..67; rows 72..79 stay zero
    // (absorb the row-group-1 waste reads). Stride 73 breaks bank conflicts.
    __shared__ float XS[80][73];
    // Per-wave row-sum buffers (rows 0..23 used; 32 so g=1 stores are in-bounds).
    __shared__ float RS[16][32][17];

    const int tid  = threadIdx.x;
    const int lane = tid & 31;
    const int w    = tid >> 5;          // wave id 0..15
    const int wr   = w >> 2;            // sub-tile row 0..3
    const int wc   = w & 3;             // sub-tile col 0..3
    const int tx = blockIdx.x;          // 64-wide tile column (0..7)
    const int ty = blockIdx.y;          // 64-tall tile row    (0..7)
    const int bc = blockIdx.z;          // b*3 + cin
    const int b   = bc / 3;
    const int cin = bc % 3;

    const int R0 = wr * 16;             // sub-tile origin inside 64x64 region
    const int C0 = wc * 16;
    const int r0g = ty * 64 + R0;       // sub-tile origin in the image
    const int c0g = tx * 64 + C0;

    // ---- zero the halo (covers borders + pad rows 72..79) ----
    for (int i = tid; i < 80 * 73; i += 512) ((float*)XS)[i] = 0.0f;
    __syncthreads();

    // ---- async global -> LDS stage of the 72x72 halo region ----
    const float* xbase = x + (size_t)bc * (H * W);
    unsigned xs_base = (unsigned)(uintptr_t)(&XS[0][0]);
    for (int e = tid; e < 72 * 72; e += 512) {
        int rr = e / 72, cc = e % 72;
        int gr = ty * 64 - 4 + rr;
        int gc = tx * 64 - 4 + cc;
        if (gr >= 0 && gr < H && gc >= 0 && gc < W) {
            unsigned loff = xs_base + (unsigned)((rr * 73 + cc) * 4);
            unsigned goff = (unsigned)((gr * W + gc) * 4);
            asm volatile("global_load_async_to_lds_b32 %0, %1, %2"
                         :: "v"(loff), "v"(goff), "s"(xbase) : "memory");
        }
    }
    asm volatile("s_wait_asynccnt 0" ::: "memory");
    __syncthreads();                    // all waves' halo data visible

    const int m16 = lane & 15;          // N (B/C/D) or M (A) index for this lane
    const int hi  = lane >> 4;          // lane-half
    const int bk  = hi * 2;             // base K of this lane's fragment elems

    // ---- channel `cin`: passthrough copy of x (streaming NT store) ----
    {
        size_t obase = (((size_t)(b * COUT + cin)) * H + r0g) * W + c0g;
        #pragma unroll
        for (int v = 0; v < 8; ++v) {
            int m = v + 8 * hi;
            __builtin_nontemporal_store(XS[4 + R0 + m][4 + C0 + m16],
                                        &out[obase + (size_t)m * W + m16]);
        }
    }

    const int RAD[3] = {1, 2, 4};       // k = 3, 5, 9
    for (int ridx = 0; ridx < 3; ++ridx) {
        const int r = RAD[ridx];
        v8f S[2];

        for (int q = 0; q < 2; ++q) {   // q=0: sum(x), q=1: sum(x^2)
            // ---- row pass: RS[w][0..23][n] = windowed column sums ----
            for (int g = 0; g < 2; ++g) {
                v8f acc;
                #pragma unroll
                for (int i = 0; i < 8; ++i) acc[i] = 0.0f;
                #pragma unroll
                for (int cch = 0; cch < 6; ++cch) {     // K = 24 in chunks of 4
                    int k0 = 4 * cch + bk;
                    v2f a;
                    a.x = XS[R0 + m16 + 16 * g][C0 + k0];
                    a.y = XS[R0 + m16 + 16 * g][C0 + k0 + 1];
                    if (q) a = a * a;                   // x^2 fragments for free
                    v2f bb;                             // banded 0/1 matrix B_r
                    bb.x = ((unsigned)(m16 + 4 - k0 + r)       <= (unsigned)(2 * r)) ? 1.0f : 0.0f;
                    bb.y = ((unsigned)(m16 + 4 - (k0 + 1) + r) <= (unsigned)(2 * r)) ? 1.0f : 0.0f;
                    acc = __builtin_amdgcn_wmma_f32_16x16x4_f32(
                        false, a, false, bb, (short)0, acc, false, false);
                }
                // D layout: vgpr v -> rows v (lanes 0-15) / v+8 (lanes 16-31)
                #pragma unroll
                for (int v = 0; v < 8; ++v) {
                    RS[w][v + 8 * hi + 16 * g][m16] = acc[v];
                }
            }

            // ---- column pass: S = C_r(16x24, banded) * RS(24x16) ----
            v8f acc;
            #pragma unroll
            for (int i = 0; i < 8; ++i) acc[i] = 0.0f;
            #pragma unroll
            for (int cch = 0; cch < 6; ++cch) {
                int k0 = 4 * cch + bk;
                v2f a;                                  // banded matrix as A
                a.x = ((unsigned)(m16 + 4 - k0 + r)       <= (unsigned)(2 * r)) ? 1.0f : 0.0f;
                a.y = ((unsigned)(m16 + 4 - (k0 + 1) + r) <= (unsigned)(2 * r)) ? 1.0f : 0.0f;
                v2f bb;                                 // R rows k0, k0+1
                bb.x = RS[w][k0][m16];
                bb.y = RS[w][k0 + 1][m16];
                acc = __builtin_amdgcn_wmma_f32_16x16x4_f32(
                    false, a, false, bb, (short)0, acc, false, false);
            }
            S[q] = acc;
        }

        // ---- variance = clip(S2/k^2 - (S1/k^2)^2, 0), NT streaming store ----
        const float inv = 1.0f / (float)((2 * r + 1) * (2 * r + 1));
        size_t obase = (((size_t)(b * COUT + 3 * (1 + ridx) + cin)) * H + r0g) * W + c0g;
        #pragma unroll
        for (int v = 0; v < 8; ++v) {
            float mean = S[0][v] * inv;
            float msq  = S[1][v] * inv;
            float var  = msq - mean * mean;
            var = var > 0.0f ? var : 0.0f;
            int m = v + 8 * hi;
            __builtin_nontemporal_store(var, &out[obase + (size_t)m * W + m16]);
        }
    }
}

extern "C" void kernel_launch(void* const* d_in, const int* in_sizes, int n_in,
                              void* d_out, int out_size, void* d_ws, size_t ws_size,
                              hipStream_t stream) {
    (void)in_sizes; (void)n_in; (void)d_ws; (void)ws_size; (void)out_size;
    const float* x = (const float*)d_in[0];
    float* out = (float*)d_out;
    dim3 grid(W / 64, H / 64, 16 * 3);  // 8 x 8 region tiles, 48 (b, cin) planes
    dim3 block(512);                    // 16 waves: cooperative halo, 1 sub-tile/wave
    varpool_wmma_kernel<<<grid, block, 0, stream>>>(x, out);
}